// SingleNCCLoss_22857815949865
// MI455X (gfx1250) — compile-verified
//
#include <hip/hip_runtime.h>

typedef __attribute__((ext_vector_type(2))) float v2f;
typedef __attribute__((ext_vector_type(8))) float v8f;

#define TW 16
#define HALO 4
#define HW 24                      // 16 + 2*4 halo
#define VOL (128*128*128)
#define NVOX (2*VOL)               // total voxels for the mean

#define S1SZ (HW*HW*TW)            // 9216 per field
#define S2SZ (HW*TW*TW)            // 6144 per field
#define S3SZ (TW*TW*TW)            // 4096 per field

// LDS layout (float offsets), phase-overlapped:
//   [0 .. 46080)      : S1[5] (stages 1-2), later S3[5] (stage 3 + final)
//   [46080 .. 76800)  : IH+JH halos (stage 1), later S2[5] (stages 2-3)
#define OFF_S1  0                  // 5 * 9216 = 46080
#define OFF_R1  46080
#define OFF_IH  OFF_R1             // 13824
#define OFF_JH  (OFF_R1 + 13824)   // 13824  (ends 73728 < 76800)
#define OFF_S2  OFF_R1             // 5 * 6144 = 30720
#define OFF_S3  0                  // 5 * 4096 overlays S1
#define OFF_RED 76800
#define SMEM_FLOATS 76801          // ~300 KB of the 320 KB WGP LDS

__device__ __forceinline__ float band(int k, int n) {
    return (k >= n && k <= n + 8) ? 1.f : 0.f;   // 9-tap all-ones window
}

__device__ __forceinline__ v8f wmma4(v2f a, v2f b, v8f c) {
    return __builtin_amdgcn_wmma_f32_16x16x4_f32(
            false, a, false, b, (short)0, c, false, false);
}

__global__ void __launch_bounds__(256)
ncc_tile_kernel(const float* __restrict__ Ii, const float* __restrict__ Ji,
                float* __restrict__ out) {
    extern __shared__ float smem[];
    float* S1 = smem + OFF_S1;
    float* IH = smem + OFF_IH;
    float* JH = smem + OFF_JH;
    float* S2 = smem + OFF_S2;
    float* S3 = smem + OFF_S3;

    const int tid  = threadIdx.x;
    const int lane = tid & 31;
    const int wave = tid >> 5;
    const int m    = lane & 15;    // A-row / B-col index per WMMA layout
    const int hi   = lane >> 4;    // upper half-wave selector

    const int blk   = blockIdx.x;
    const int batch = blk >> 9;
    const int tile  = blk & 511;           // 8x8x8 tiles of 16^3
    const int tx = tile & 7, ty = (tile >> 3) & 7, tz = tile >> 6;
    const int d0 = tz * TW, h0 = ty * TW, w0 = tx * TW;

    const float* Ib = Ii + batch * VOL;
    const float* Jb = Ji + batch * VOL;

    // ---- load 24^3 halo (zero padded) for both inputs into LDS ----
    for (int idx = tid; idx < HW*HW*HW; idx += 256) {
        int dz  = idx / (HW*HW);
        int rem = idx - dz * HW*HW;
        int dy  = rem / HW;
        int dx  = rem - dy * HW;
        int gd = d0 + dz - HALO, gh = h0 + dy - HALO, gw = w0 + dx - HALO;
        float iv = 0.f, jv = 0.f;
        if ((unsigned)gd < 128u && (unsigned)gh < 128u && (unsigned)gw < 128u) {
            int g = (gd * 128 + gh) * 128 + gw;
            iv = Ib[g];
            jv = Jb[g];
        }
        IH[idx] = iv;
        JH[idx] = jv;
    }
    if (tid == 0) smem[OFF_RED] = 0.f;
    __syncthreads();

    // ---- Stage 1: sum along W for all 5 fields at once.
    //      rows = (dz,dy) in [0,576) -> 36 M-tiles; shared IH/JH loads,
    //      5 independent WMMA accumulator chains per k-step. ----
    for (int t = wave; t < 36; t += 8) {
        int row   = t * TW + m;
        int adz   = row / HW;
        int ady   = row - adz * HW;
        int abase = (adz * HW + ady) * HW;
        v8f c0 = {}, c1 = {}, c2 = {}, c3 = {}, c4 = {};
        #pragma unroll
        for (int k0 = 0; k0 < HW; k0 += 4) {
            int ka = k0 + hi * 2;
            float i0 = IH[abase + ka], i1 = IH[abase + ka + 1];
            float j0 = JH[abase + ka], j1 = JH[abase + ka + 1];
            v2f b;                              // constant band fragment
            b.x = band(ka, m);
            b.y = band(ka + 1, m);
            v2f a;
            a.x = i0;       a.y = i1;       c0 = wmma4(a, b, c0);
            a.x = j0;       a.y = j1;       c1 = wmma4(a, b, c1);
            a.x = i0 * i0;  a.y = i1 * i1;  c2 = wmma4(a, b, c2);
            a.x = j0 * j0;  a.y = j1 * j1;  c3 = wmma4(a, b, c3);
            a.x = i0 * j0;  a.y = i1 * j1;  c4 = wmma4(a, b, c4);
        }
        #pragma unroll
        for (int v = 0; v < 8; ++v) {
            int orow = t * TW + v + hi * 8;
            int odz  = orow / HW;
            int ody  = orow - odz * HW;
            int o    = (odz * HW + ody) * TW + m;
            S1[o]          = c0[v];
            S1[S1SZ   + o] = c1[v];
            S1[2*S1SZ + o] = c2[v];
            S1[3*S1SZ + o] = c3[v];
            S1[4*S1SZ + o] = c4[v];
        }
    }
    __syncthreads();   // halos dead; R1 becomes S2[5]

    // ---- Stage 2: sum along H. One C-tile per dz slice, 5 fields fused. ----
    for (int t = wave; t < HW; t += 8) {
        v8f c0 = {}, c1 = {}, c2 = {}, c3 = {}, c4 = {};
        #pragma unroll
        for (int k0 = 0; k0 < HW; k0 += 4) {
            int ka = k0 + hi * 2;
            v2f a;                              // A[m][k] = band(k,m)
            a.x = band(ka, m);
            a.y = band(ka + 1, m);
            int o0 = (t * HW + ka) * TW + m;    // S1[dz][k][n]
            int o1 = o0 + TW;                   // k+1 row
            v2f b;
            b.x = S1[o0];          b.y = S1[o1];          c0 = wmma4(a, b, c0);
            b.x = S1[S1SZ   + o0]; b.y = S1[S1SZ   + o1]; c1 = wmma4(a, b, c1);
            b.x = S1[2*S1SZ + o0]; b.y = S1[2*S1SZ + o1]; c2 = wmma4(a, b, c2);
            b.x = S1[3*S1SZ + o0]; b.y = S1[3*S1SZ + o1]; c3 = wmma4(a, b, c3);
            b.x = S1[4*S1SZ + o0]; b.y = S1[4*S1SZ + o1]; c4 = wmma4(a, b, c4);
        }
        #pragma unroll
        for (int v = 0; v < 8; ++v) {
            int y = v + hi * 8;
            int o = (t * TW + y) * TW + m;      // S2[dz][y][x]
            S2[o]          = c0[v];
            S2[S2SZ   + o] = c1[v];
            S2[2*S2SZ + o] = c2[v];
            S2[3*S2SZ + o] = c3[v];
            S2[4*S2SZ + o] = c4[v];
        }
    }
    __syncthreads();   // S1 dead; R0 becomes S3[5]

    // ---- Stage 3: sum along D. One C-tile per y slice, 5 fields fused. ----
    for (int t = wave; t < TW; t += 8) {
        v8f c0 = {}, c1 = {}, c2 = {}, c3 = {}, c4 = {};
        #pragma unroll
        for (int k0 = 0; k0 < HW; k0 += 4) {
            int ka = k0 + hi * 2;
            v2f a;
            a.x = band(ka, m);
            a.y = band(ka + 1, m);
            int o0 = (ka * TW + t) * TW + m;    // S2[k][y][n]
            int o1 = o0 + TW * TW;              // k+1 slice
            v2f b;
            b.x = S2[o0];          b.y = S2[o1];          c0 = wmma4(a, b, c0);
            b.x = S2[S2SZ   + o0]; b.y = S2[S2SZ   + o1]; c1 = wmma4(a, b, c1);
            b.x = S2[2*S2SZ + o0]; b.y = S2[2*S2SZ + o1]; c2 = wmma4(a, b, c2);
            b.x = S2[3*S2SZ + o0]; b.y = S2[3*S2SZ + o1]; c3 = wmma4(a, b, c3);
            b.x = S2[4*S2SZ + o0]; b.y = S2[4*S2SZ + o1]; c4 = wmma4(a, b, c4);
        }
        #pragma unroll
        for (int v = 0; v < 8; ++v) {
            int z = v + hi * 8;
            int o = (z * TW + t) * TW + m;      // S3[z][y][x]
            S3[o]          = c0[v];
            S3[S3SZ   + o] = c1[v];
            S3[2*S3SZ + o] = c2[v];
            S3[3*S3SZ + o] = c3[v];
            S3[4*S3SZ + o] = c4[v];
        }
    }
    __syncthreads();

    // ---- per-voxel cc and block reduction ----
    const float inv = 1.f / 729.f;
    float local = 0.f;
    for (int idx = tid; idx < TW*TW*TW; idx += 256) {
        float Is  = S3[idx];
        float Js  = S3[S3SZ   + idx];
        float I2s = S3[2*S3SZ + idx];
        float J2s = S3[3*S3SZ + idx];
        float IJs = S3[4*S3SZ + idx];
        float cross = IJs - Is * Js * inv;
        float Ivar  = I2s - Is * Is * inv;
        float Jvar  = J2s - Js * Js * inv;
        local += cross * cross / (Ivar * Jvar + 1e-5f);
    }
    atomicAdd(&smem[OFF_RED], local);   // ds_add_f32
    __syncthreads();
    if (tid == 0) {
        atomicAdd(out, -smem[OFF_RED] * (1.0f / (float)NVOX));
    }
}

__global__ void zero_out_kernel(float* out) {
    out[0] = 0.f;
}

extern "C" void kernel_launch(void* const* d_in, const int* in_sizes, int n_in,
                              void* d_out, int out_size, void* d_ws, size_t ws_size,
                              hipStream_t stream) {
    (void)in_sizes; (void)n_in; (void)out_size; (void)d_ws; (void)ws_size;
    const float* Ii = (const float*)d_in[0];
    const float* Ji = (const float*)d_in[1];
    float* out = (float*)d_out;

    size_t smemBytes = (size_t)SMEM_FLOATS * sizeof(float);
    (void)hipFuncSetAttribute((const void*)ncc_tile_kernel,
                              hipFuncAttributeMaxDynamicSharedMemorySize,
                              (int)smemBytes);

    zero_out_kernel<<<1, 1, 0, stream>>>(out);
    // 2 batches * 8^3 tiles of 16^3 = 1024 workgroups of 8 wave32s
    ncc_tile_kernel<<<1024, 256, smemBytes, stream>>>(Ii, Ji, out);
}